// SphericalExpansion_40561671144110
// MI455X (gfx1250) — compile-verified
//
#include <hip/hip_runtime.h>
#include <hip/hip_bf16.h>

typedef __attribute__((ext_vector_type(2))) float v2f;
typedef __attribute__((ext_vector_type(4))) float v4f;
typedef __attribute__((ext_vector_type(8))) float v8f;

#define NCENTERS 16384
#define OUTROW 512

// ---------------- pass 0: zero histogram ----------------
__global__ void zero_kernel(int* __restrict__ p, int n) {
    int i = blockIdx.x * blockDim.x + threadIdx.x;
    if (i < n) p[i] = 0;
}

// ---------------- pass 1: histogram of density_indices ----------------
__global__ void hist_kernel(const int* __restrict__ di, int* __restrict__ counts, int P) {
    int i = blockIdx.x * blockDim.x + threadIdx.x;
    if (i < P) atomicAdd(&counts[di[i]], 1);
}

// ---------------- pass 2: exclusive scan of 16384 bins (1 block) ----------------
__global__ void scan_kernel(const int* __restrict__ counts, int* __restrict__ offsets,
                            int* __restrict__ cursors, int P, int C) {
    __shared__ int buf[2][1024];
    const int t = threadIdx.x;             // 0..1023
    const int per = C / 1024;              // 16 bins per thread
    const int base = t * per;
    int s = 0;
    for (int i = 0; i < per; ++i) s += counts[base + i];
    buf[0][t] = s;
    __syncthreads();
    int src = 0;
    for (int off = 1; off < 1024; off <<= 1) {
        int v = buf[src][t];
        if (t >= off) v += buf[src][t - off];
        buf[src ^ 1][t] = v;
        src ^= 1;
        __syncthreads();
    }
    int run = buf[src][t] - s;             // exclusive prefix for this thread's chunk
    for (int i = 0; i < per; ++i) {
        offsets[base + i] = run;
        cursors[base + i] = run;
        run += counts[base + i];
    }
    if (t == 0) offsets[C] = P;
}

// ---------------- pass 3: scatter neighbor ids into per-center buckets ----------------
__global__ void scatter_kernel(const int* __restrict__ di, int* __restrict__ cursors,
                               int* __restrict__ order, int P) {
    int i = blockIdx.x * blockDim.x + threadIdx.x;
    if (i < P) {
        int pos = atomicAdd(&cursors[di[i]], 1);
        order[pos] = i;
    }
}

// ---------------- pass 4: per-center dense reduction via WMMA f32 16x16x4 ----------------
// One wave (32 threads) per center. For its K neighbors (steps of 4):
//   A_q[lm,k] = sph[p_k, lm] * W[q, species[p_k]]   (16 x 4)
//   B[k,col]  = radial_basis[p_k, col], col=(l',n)  (4 x 32 -> two 16-wide N tiles)
//   D_q += A_q * B ; keep cols with l' == l(lm); out[lm*32 + q*8 + n]
__global__ __launch_bounds__(32) void spexp_wmma_kernel(
    const float* __restrict__ rb,      // (P, 32)
    const float* __restrict__ sph,     // (P, 16)
    const float* __restrict__ W,       // (4, 4) row-major: W[q*4 + s]
    const int*   __restrict__ species, // (P,)
    const int*   __restrict__ order,   // (P,) sorted neighbor ids
    const int*   __restrict__ offsets, // (C+1,)
    float*       __restrict__ out)     // (C, 512)
{
    __shared__ alignas(16) float sW[16];
    __shared__ alignas(16) float sS[4][16];   // sph rows for 4 neighbors
    __shared__ alignas(16) float sR[4][32];   // rb rows for 4 neighbors
    __shared__ alignas(16) float sP[4][4];    // pseudo[k][q]

    const int c    = blockIdx.x;
    const int lane = threadIdx.x;
    const int half = lane >> 4;    // 0: lanes 0-15, 1: lanes 16-31
    const int idx  = lane & 15;

    if (lane < 16) sW[lane] = W[lane];

    const int start = offsets[c];
    const int end   = offsets[c + 1];
    const int count = end - start;
    const int nfull = count & ~3;   // neighbors handled by unpredicated loop
    const int tail  = count & 3;

    v8f acc[4][2];
    const v8f vzero = {0.f, 0.f, 0.f, 0.f, 0.f, 0.f, 0.f, 0.f};
#pragma unroll
    for (int q = 0; q < 4; ++q)
#pragma unroll
        for (int nt = 0; nt < 2; ++nt) acc[q][nt] = vzero;

    __syncthreads();   // sW visible

    const int k = lane >> 3;   // neighbor slot 0..3 this lane stages
    const int j = lane & 7;    // element-group within slot
    const int k0 = half * 2;   // this half-wave's K pair (ISA A/B layout)

    auto do_wmma = [&]() {
#pragma unroll
        for (int q = 0; q < 4; ++q) {
            v2f a;
            a.x = sS[k0][idx]     * sP[k0][q];
            a.y = sS[k0 + 1][idx] * sP[k0 + 1][q];
#pragma unroll
            for (int nt = 0; nt < 2; ++nt) {
                const int col = nt * 16 + idx;
                v2f b;
                b.x = sR[k0][col];
                b.y = sR[k0 + 1][col];
                acc[q][nt] = __builtin_amdgcn_wmma_f32_16x16x4_f32(
                    false, a, false, b, (short)0, acc[q][nt], false, false);
            }
        }
    };

    // ---- main loop: full groups of 4 neighbors, fully unpredicated staging ----
    int basep = start;
    for (; basep < start + nfull; basep += 4) {
        const int p = order[basep + k];
        if (j < 4) {
            v4f v = __builtin_nontemporal_load(((const v4f*)(sph + (size_t)p * 16)) + j);
            ((v4f*)&sS[k][0])[j] = v;
        }
        {
            v4f v = __builtin_nontemporal_load(((const v4f*)(rb + (size_t)p * 32)) + j);
            ((v4f*)&sR[k][0])[j] = v;
        }
        if (j < 4) sP[k][j] = sW[j * 4 + species[p]];
        __syncthreads();
        do_wmma();
        __syncthreads();
    }

    // ---- tail: zero-fill LDS, then overwrite valid slots under exec mask ----
    if (tail) {
        const v4f z4 = {0.f, 0.f, 0.f, 0.f};
        if (j < 4) ((v4f*)&sS[k][0])[j] = z4;
        ((v4f*)&sR[k][0])[j] = z4;
        if (j < 4) sP[k][j] = 0.f;
        const bool valid = k < tail;
        if (valid) {
            const int p = order[basep + k];
            if (j < 4) {
                v4f v = __builtin_nontemporal_load(((const v4f*)(sph + (size_t)p * 16)) + j);
                ((v4f*)&sS[k][0])[j] = v;
            }
            {
                v4f v = __builtin_nontemporal_load(((const v4f*)(rb + (size_t)p * 32)) + j);
                ((v4f*)&sR[k][0])[j] = v;
            }
            if (j < 4) sP[k][j] = sW[j * 4 + species[p]];
        }
        __syncthreads();
        do_wmma();
    }

    // ---- write 512 outputs (each element written exactly once, no atomics) ----
    float* outc = out + (size_t)c * OUTROW;
#pragma unroll
    for (int q = 0; q < 4; ++q)
#pragma unroll
        for (int nt = 0; nt < 2; ++nt)
#pragma unroll
            for (int r = 0; r < 8; ++r) {
                const int M = r + 8 * half;                       // lm = l^2 + m
                const int l = (M >= 9) ? 3 : (M >= 4) ? 2 : (M >= 1) ? 1 : 0;
                const int col = nt * 16 + idx;                    // (l', n)
                if ((col >> 3) == l)
                    outc[M * 32 + q * 8 + (col & 7)] = acc[q][nt][r];
            }
}

extern "C" void kernel_launch(void* const* d_in, const int* in_sizes, int n_in,
                              void* d_out, int out_size, void* d_ws, size_t ws_size,
                              hipStream_t stream) {
    const float* rb      = (const float*)d_in[0];   // radial_basis (P,4,8)
    const float* sph     = (const float*)d_in[1];   // sph (P,16)
    const float* W       = (const float*)d_in[2];   // (4,4)
    const int*   species = (const int*)d_in[3];     // (P,)
    const int*   di      = (const int*)d_in[4];     // density_indices (P,)
    const int    P       = in_sizes[3];
    const int    C       = NCENTERS;
    float*       out     = (float*)d_out;

    // workspace layout (ints): counts[C] | offsets[C+1] | cursors[C] | order[P]
    int* counts  = (int*)d_ws;
    int* offsets = counts + C;
    int* cursors = offsets + C + 1;
    int* order   = cursors + C;

    zero_kernel<<<(C + 255) / 256, 256, 0, stream>>>(counts, C);
    hist_kernel<<<(P + 255) / 256, 256, 0, stream>>>(di, counts, P);
    scan_kernel<<<1, 1024, 0, stream>>>(counts, offsets, cursors, P, C);
    scatter_kernel<<<(P + 255) / 256, 256, 0, stream>>>(di, cursors, order, P);
    spexp_wmma_kernel<<<C, 32, 0, stream>>>(rb, sph, W, species, order, offsets, out);
}